// TransformerQQ_56332791054968
// MI455X (gfx1250) — compile-verified
//
#include <hip/hip_runtime.h>

typedef __attribute__((ext_vector_type(2))) float v2f;
typedef __attribute__((ext_vector_type(4))) float v4f;
typedef __attribute__((ext_vector_type(8))) float v8f;
typedef __attribute__((ext_vector_type(4))) int   v4i;

#if defined(__AMDGCN__) && __has_builtin(__builtin_amdgcn_global_load_async_to_lds_b128)
#define USE_ASYNC_LDS 1
#else
#define USE_ASYNC_LDS 0
#endif

#define B_ 8
#define H_ 32
#define C_ 2048
#define D_ 128
#define O_ 32
#define ALPHA_ 0.01f
#define CT_ 16           // tokens per workgroup
#define ROWP_ 132        // padded LDS row stride in floats (132 % 64 = 4 -> conflict-free)

// LDS layout (float offsets)
#define XS_OFF   0
#define XS_SZ    (H_ * CT_ * ROWP_)        // 67584 floats
#define W_OFF    (XS_OFF + XS_SZ)
#define W_SZ     (O_ * ROWP_)              // 4224 floats
#define TOP_OFF  (W_OFF + W_SZ)
#define TOP_SZ   (CT_ * H_ * 4)            // v4f per (m,h)
#define AUXP_OFF (TOP_OFF + TOP_SZ)
#define AUXP_SZ  O_
#define CNT_OFF  (AUXP_OFF + AUXP_SZ)
#define CNT_SZ   O_
#define LDS_FLOATS (CNT_OFF + CNT_SZ)      // ~73920 floats = ~289 KB <= 320 KB

#if USE_ASYNC_LDS
#define AS1_ __attribute__((address_space(1)))
#define AS3_ __attribute__((address_space(3)))
__device__ static inline void async_copy16(const float* g, float* l) {
  AS1_ v4i* gp = (AS1_ v4i*)g;   // global (as1) source
  AS3_ v4i* lp = (AS3_ v4i*)l;   // LDS (as3) destination
  __builtin_amdgcn_global_load_async_to_lds_b128(gp, lp, 0, 0);
}
#endif

__global__ void __launch_bounds__(256, 1)
moe_router_kernel(const float* __restrict__ x, const float* __restrict__ W,
                  float* __restrict__ y, float* __restrict__ gprobs,
                  float* __restrict__ gcnts) {
  extern __shared__ float lds[];
  float* xs   = lds + XS_OFF;
  float* wl   = lds + W_OFF;
  v4f*   tops = (v4f*)(lds + TOP_OFF);
  float* auxp = lds + AUXP_OFF;
  float* cnts = lds + CNT_OFF;

  const int tid  = threadIdx.x;
  const int lane = tid & 31;
  const int wave = tid >> 5;
  const int b    = blockIdx.x >> 7;           // 128 c-tiles per batch
  const int c0   = (blockIdx.x & 127) * CT_;

  // zero the small per-block accumulators (auxp and cnts are contiguous)
  for (int i = tid; i < AUXP_SZ + CNT_SZ; i += 256) auxp[i] = 0.0f;

  // ---- Phase 1: stage x tile (all 32 heads x 16 tokens) and W into LDS ----
#if USE_ASYNC_LDS
  for (int q = tid; q < H_ * CT_ * (D_ / 4); q += 256) {
    int row = q >> 5;             // row = h*16 + m
    int d4  = q & 31;
    int h = row >> 4, m = row & 15;
    async_copy16(x + (size_t)((b * H_ + h) * C_ + c0 + m) * D_ + d4 * 4,
                 &xs[row * ROWP_ + d4 * 4]);
  }
  for (int q = tid; q < O_ * (D_ / 4); q += 256) {
    int o = q >> 5, d4 = q & 31;
    async_copy16(W + o * D_ + d4 * 4, &wl[o * ROWP_ + d4 * 4]);
  }
#if __has_builtin(__builtin_amdgcn_s_wait_asynccnt)
  __builtin_amdgcn_s_wait_asynccnt(0);
#else
  asm volatile("s_wait_asynccnt 0" ::: "memory");
#endif
#else
  const v4f* xg = (const v4f*)x;
  for (int q = tid; q < H_ * CT_ * (D_ / 4); q += 256) {
    int row = q >> 5;
    int d4  = q & 31;
    int h = row >> 4, m = row & 15;
    v4f v = __builtin_nontemporal_load(
        xg + ((size_t)((b * H_ + h) * C_ + c0 + m) * D_ >> 2) + d4);
    *(v4f*)&xs[row * ROWP_ + d4 * 4] = v;
  }
  const v4f* wg = (const v4f*)W;
  for (int q = tid; q < O_ * (D_ / 4); q += 256) {
    int o = q >> 5, d4 = q & 31;
    *(v4f*)&wl[o * ROWP_ + d4 * 4] = wg[(o * D_ >> 2) + d4];
  }
#endif
  __syncthreads();

  // ---- Phase 2: WMMA logits (K-outer, 4 heads share B fragments) ----
  const int l16  = lane & 15;
  const int hi16 = lane >> 4;     // 0 or 1
  const int koff = hi16 * 2;      // f32 A/B fragment: lanes>=16 hold K+2,K+3
  float auxa = 0.0f, auxb = 0.0f;

  const float* b0row = &wl[l16 * ROWP_];
  const float* b1row = &wl[(l16 + 16) * ROWP_];
  const float* ar0 = &xs[((wave     ) * CT_ + l16) * ROWP_];
  const float* ar1 = &xs[((wave +  8) * CT_ + l16) * ROWP_];
  const float* ar2 = &xs[((wave + 16) * CT_ + l16) * ROWP_];
  const float* ar3 = &xs[((wave + 24) * CT_ + l16) * ROWP_];

  v8f acc[8] = {};   // [head-pair: 4 heads x 2 O-tiles]
  #pragma unroll 4
  for (int k0 = 0; k0 < D_; k0 += 4) {
    const int kk = k0 + koff;
    v2f b0, b1, a;
    b0.x = b0row[kk]; b0.y = b0row[kk + 1];
    b1.x = b1row[kk]; b1.y = b1row[kk + 1];
    a.x = ar0[kk]; a.y = ar0[kk + 1];
    acc[0] = __builtin_amdgcn_wmma_f32_16x16x4_f32(false, a, false, b0, (short)0, acc[0], false, false);
    acc[1] = __builtin_amdgcn_wmma_f32_16x16x4_f32(false, a, false, b1, (short)0, acc[1], false, false);
    a.x = ar1[kk]; a.y = ar1[kk + 1];
    acc[2] = __builtin_amdgcn_wmma_f32_16x16x4_f32(false, a, false, b0, (short)0, acc[2], false, false);
    acc[3] = __builtin_amdgcn_wmma_f32_16x16x4_f32(false, a, false, b1, (short)0, acc[3], false, false);
    a.x = ar2[kk]; a.y = ar2[kk + 1];
    acc[4] = __builtin_amdgcn_wmma_f32_16x16x4_f32(false, a, false, b0, (short)0, acc[4], false, false);
    acc[5] = __builtin_amdgcn_wmma_f32_16x16x4_f32(false, a, false, b1, (short)0, acc[5], false, false);
    a.x = ar3[kk]; a.y = ar3[kk + 1];
    acc[6] = __builtin_amdgcn_wmma_f32_16x16x4_f32(false, a, false, b0, (short)0, acc[6], false, false);
    acc[7] = __builtin_amdgcn_wmma_f32_16x16x4_f32(false, a, false, b1, (short)0, acc[7], false, false);
  }

  // ---- Softmax + top2 per token row (cross-lane within 16-lane halves) ----
  // C layout: VGPR r holds token row M=r (lanes 0-15) / M=r+8 (lanes 16-31),
  // col = lane%16 (tile0 -> o 0..15, tile1 -> o 16..31)
  #pragma unroll
  for (int hh = 0; hh < 4; ++hh) {
    const int h = wave + hh * 8;
    #pragma unroll
    for (int r = 0; r < 8; ++r) {
      float v0 = acc[2 * hh][r], v1 = acc[2 * hh + 1][r];
      float hv, lv; int hidx, lidx;
      if (v0 >= v1) { hv = v0; hidx = l16;      lv = v1; lidx = l16 + 16; }
      else          { hv = v1; hidx = l16 + 16; lv = v0; lidx = l16;      }
      #pragma unroll
      for (int mask = 1; mask <= 8; mask <<= 1) {   // stays within 16-lane half
        float ohv = __shfl_xor(hv, mask, 32);
        float olv = __shfl_xor(lv, mask, 32);
        int   ohi = __shfl_xor(hidx, mask, 32);
        int   oli = __shfl_xor(lidx, mask, 32);
        if (ohv > hv) {
          float ns = (hv >= olv) ? hv : olv;
          int  nsi = (hv >= olv) ? hidx : oli;
          hv = ohv; hidx = ohi; lv = ns; lidx = nsi;
        } else {
          float ns = (ohv >= lv) ? ohv : lv;
          int  nsi = (ohv >= lv) ? ohi : lidx;
          lv = ns; lidx = nsi;
        }
      }
      const float mx = hv;                       // row max
      float e0 = __expf(v0 - mx), e1 = __expf(v1 - mx);
      float s = e0 + e1;
      #pragma unroll
      for (int mask = 1; mask <= 8; mask <<= 1) s += __shfl_xor(s, mask, 32);
      const float inv = 1.0f / s;
      auxa += e0 * inv;                          // probs for aux mean
      auxb += e1 * inv;
      if (l16 == 0) {
        const int m = r + hi16 * 8;
        const float p1 = __expf(lv - mx) * inv;
        v4f t; t.x = inv; t.y = p1; t.z = (float)hidx; t.w = (float)lidx;
        tops[m * H_ + h] = t;                    // p0 = exp(hv-mx)/s = inv
        atomicAdd(&cnts[hidx], 1.0f);
        atomicAdd(&cnts[lidx], 1.0f);
      }
    }
  }
  atomicAdd(&auxp[l16], auxa);
  atomicAdd(&auxp[l16 + 16], auxb);
  __syncthreads();

  // ---- Phase 3: cross-head gather (LDS only) + weighted sum -> y (NT store) ----
  v4f* yg = (v4f*)y;
  const v4f* xs4 = (const v4f*)xs;               // 528B rows -> 16B aligned
  for (int q = tid; q < CT_ * H_ * (D_ / 4); q += 256) {
    int row = q >> 5;              // row = m*32 + h (uniform per wave iteration)
    int d4  = q & 31;
    int m = row >> 5, h = row & 31;
    v4f t = tops[row];
    int i0 = (int)t.z, i1 = (int)t.w;
    v4f s0 = xs4[(i0 * CT_ + m) * (ROWP_ / 4) + d4];
    v4f s1 = xs4[(i1 * CT_ + m) * (ROWP_ / 4) + d4];
    v4f o = s0 * t.x + s1 * t.y;
    __builtin_nontemporal_store(
        o, yg + ((size_t)((b * H_ + h) * C_ + c0 + m) * D_ >> 2) + d4);
  }

  // ---- Phase 4: flush aux accumulators ----
  if (tid < O_) {
    atomicAdd(&gprobs[tid], auxp[tid]);
    atomicAdd(&gcnts[tid], cnts[tid]);
  }
}

__global__ void moe_ws_init(float* ws) { ws[threadIdx.x] = 0.0f; }

__global__ void moe_aux_finalize(const float* __restrict__ gprobs,
                                 const float* __restrict__ gcnts,
                                 float* __restrict__ aux_out) {
  const int lane = threadIdx.x;                  // 32 threads
  float p = gprobs[lane];
  float c = gcnts[lane];
  float ctot = c;
  #pragma unroll
  for (int mask = 1; mask <= 16; mask <<= 1) ctot += __shfl_xor(ctot, mask, 32);
  const float ntok = (float)(B_ * H_ * C_);
  float term = (p / ntok) * (c / ctot);
  #pragma unroll
  for (int mask = 1; mask <= 16; mask <<= 1) term += __shfl_xor(term, mask, 32);
  if (lane == 0) *aux_out = ALPHA_ * (float)O_ * term;
}

extern "C" void kernel_launch(void* const* d_in, const int* in_sizes, int n_in,
                              void* d_out, int out_size, void* d_ws, size_t ws_size,
                              hipStream_t stream) {
  const float* x = (const float*)d_in[0];
  const float* W = (const float*)d_in[1];
  float* y   = (float*)d_out;
  float* aux = y + (size_t)B_ * H_ * C_ * D_;
  float* gprobs = (float*)d_ws;
  float* gcnts  = gprobs + O_;

  moe_ws_init<<<1, 2 * O_, 0, stream>>>(gprobs);
  dim3 grid(B_ * (C_ / CT_));                    // 1024 workgroups
  moe_router_kernel<<<grid, 256, LDS_FLOATS * sizeof(float), stream>>>(
      x, W, y, gprobs, gcnts);
  moe_aux_finalize<<<1, O_, 0, stream>>>(gprobs, gcnts, aux);
}